// TestFP8_QDQ_14431090115263
// MI455X (gfx1250) — compile-verified
//
#include <hip/hip_runtime.h>
#include <hip/hip_bf16.h>

// FP8 E4M3 (OCP) max finite magnitude — matches jnp.float8_e4m3fn.
#define FP8_MAX_E4M3 448.0f

typedef __attribute__((ext_vector_type(4))) float v4f;
typedef __attribute__((ext_vector_type(2))) float v2f;

// Process one float4: update running amax, QDQ through hardware E4M3 (RNE).
__device__ __forceinline__ void qdq_vec4(const v4f* __restrict__ in4,
                                         v4f* __restrict__ out4,
                                         int i, float s, float& lmax) {
    v4f v = __builtin_nontemporal_load(&in4[i]);

    // amax of the raw input
    lmax = fmaxf(lmax, fabsf(v.x));
    lmax = fmaxf(lmax, fabsf(v.y));
    lmax = fmaxf(lmax, fabsf(v.z));
    lmax = fmaxf(lmax, fabsf(v.w));

    // clip(inp*scale, +-448) exactly as the reference, then HW E4M3 RNE cast
    float cx = fminf(fmaxf(v.x * s, -FP8_MAX_E4M3), FP8_MAX_E4M3);
    float cy = fminf(fmaxf(v.y * s, -FP8_MAX_E4M3), FP8_MAX_E4M3);
    float cz = fminf(fmaxf(v.z * s, -FP8_MAX_E4M3), FP8_MAX_E4M3);
    float cw = fminf(fmaxf(v.w * s, -FP8_MAX_E4M3), FP8_MAX_E4M3);

    int p01 = __builtin_amdgcn_cvt_pk_fp8_f32(cx, cy, 0, false); // v_cvt_pk_fp8_f32
    int p23 = __builtin_amdgcn_cvt_pk_fp8_f32(cz, cw, 0, false);
    v2f r01 = __builtin_amdgcn_cvt_pk_f32_fp8(p01, false);        // v_cvt_pk_f32_fp8
    v2f r23 = __builtin_amdgcn_cvt_pk_f32_fp8(p23, false);

    v4f o;
    o.x = r01.x; o.y = r01.y; o.z = r23.x; o.w = r23.y;
    __builtin_nontemporal_store(o, &out4[i]);
}

// ---------------------------------------------------------------------------
// Main streaming QDQ kernel (defined first so the disasm snippet shows it).
//   - 128-bit non-temporal loads/stores (512 MB streamed once; > 192 MB L2)
//   - 2x unrolled grid-stride loop: two independent b128 loads in flight
//   - hardware E4M3 round trip: v_cvt_pk_fp8_f32 -> v_cvt_pk_f32_fp8 (RNE)
//   - amax: thread fmax -> wave32 shfl_xor butterfly -> LDS -> 1 atomic/block
// ---------------------------------------------------------------------------
__global__ void fp8_qdq_kernel(const v4f* __restrict__ in4,
                               v4f* __restrict__ out4,
                               const float* __restrict__ scale,
                               float* __restrict__ amax_out,
                               int nvec,            // number of float4 vectors
                               const float* __restrict__ in_tail,
                               float* __restrict__ out_tail,
                               int ntail) {         // leftover scalars (0..3)
    const float s = scale[0];   // uniform scalar load (s_load, cached)

    float lmax = 0.0f;
    const int stride = gridDim.x * blockDim.x;
    const int idx = blockIdx.x * blockDim.x + threadIdx.x;

    int i = idx;
    // 2x unroll: both accesses coalesced per wave, one grid-stride apart,
    // giving two outstanding 128-bit loads before the first store waits.
    for (; i + stride < nvec; i += 2 * stride) {
        qdq_vec4(in4, out4, i, s, lmax);
        qdq_vec4(in4, out4, i + stride, s, lmax);
    }
    if (i < nvec) qdq_vec4(in4, out4, i, s, lmax);

    // Scalar tail (N not divisible by 4). ntail is 0 for 8192x8192.
    if (idx < ntail) {
        float x = in_tail[idx];
        lmax = fmaxf(lmax, fabsf(x));
        float c = fminf(fmaxf(x * s, -FP8_MAX_E4M3), FP8_MAX_E4M3);
        int p = __builtin_amdgcn_cvt_pk_fp8_f32(c, c, 0, false);
        v2f r = __builtin_amdgcn_cvt_pk_f32_fp8(p, false);
        out_tail[idx] = r.x;
    }

    // ---- amax reduction: wave32 butterfly ----
    #pragma unroll
    for (int off = 16; off > 0; off >>= 1)
        lmax = fmaxf(lmax, __shfl_xor(lmax, off, 32));

    __shared__ float smax[32];                 // up to 32 waves per block
    const int lane = threadIdx.x & 31;
    const int wave = threadIdx.x >> 5;
    if (lane == 0) smax[wave] = lmax;
    __syncthreads();

    if (threadIdx.x == 0) {
        const int nwaves = (blockDim.x + 31) >> 5;
        float bmax = smax[0];
        for (int w = 1; w < nwaves; ++w) bmax = fmaxf(bmax, smax[w]);
        // bmax >= 0, so fp32 bit pattern orders as u32 -> global_atomic_max_u32
        atomicMax(reinterpret_cast<unsigned int*>(amax_out),
                  __float_as_uint(bmax));
    }
}

// ---------------------------------------------------------------------------
// Init: amax slot = 0, scale_inv slot = 1/scale. Runs before the main kernel
// on the same stream, so ordering is guaranteed inside the captured graph.
// ---------------------------------------------------------------------------
__global__ void fp8_qdq_init_kernel(const float* __restrict__ scale,
                                    float* __restrict__ amax_out,
                                    float* __restrict__ scale_inv_out) {
    if (blockIdx.x == 0 && threadIdx.x == 0) {
        *amax_out = 0.0f;               // u32 bit pattern 0 -> atomicMax identity
        *scale_inv_out = 1.0f / scale[0];
    }
}

// ---------------------------------------------------------------------------
// Launch
// ---------------------------------------------------------------------------
extern "C" void kernel_launch(void* const* d_in, const int* in_sizes, int n_in,
                              void* d_out, int out_size, void* d_ws, size_t ws_size,
                              hipStream_t stream) {
    (void)n_in; (void)out_size; (void)d_ws; (void)ws_size;

    const float* inp   = reinterpret_cast<const float*>(d_in[0]);
    const float* scale = reinterpret_cast<const float*>(d_in[1]);
    float* out = reinterpret_cast<float*>(d_out);

    const int n = in_sizes[0];            // 8192*8192 = 67,108,864
    float* amax_slot = out + n;           // output layout: [out, amax, scale_inv]
    float* sinv_slot = out + n + 1;

    fp8_qdq_init_kernel<<<1, 32, 0, stream>>>(scale, amax_slot, sinv_slot);

    const int nvec  = n >> 2;             // float4 count
    const int ntail = n & 3;
    const float* in_tail = inp + (nvec << 2);
    float*       out_tail = out + (nvec << 2);

    const int threads = 256;              // 8 wave32 per block
    int blocks = (nvec + threads - 1) / threads;
    if (blocks > 4096) blocks = 4096;     // grid-stride; ~16 vec4 iters/thread
    if (blocks < 1) blocks = 1;

    fp8_qdq_kernel<<<blocks, threads, 0, stream>>>(
        reinterpret_cast<const v4f*>(inp),
        reinterpret_cast<v4f*>(out),
        scale, amax_slot, nvec, in_tail, out_tail, ntail);
}